// SelfAttention_6906307412484
// MI455X (gfx1250) — compile-verified
//
#include <hip/hip_runtime.h>

// ---------------------------------------------------------------------------
// Self-attention with relative_key_query positional bias, fp32, MI455X/gfx1250.
// All GEMMs on V_WMMA_F32_16X16X4_F32 (wave32 WMMA, fp32 in/out).
//
// Roofline: probs output is 201 MB; the naive pipeline moves it through HBM
// 4x (raw write, softmax r/w, P.V read) ~= 850 MB ~= 36 us @ 23.3 TB/s.
// Compute is ~53 GFLOP (trivial on the WMMA pipe), so this version fuses
// scores+softmax+P.V into one kernel and RECOMPUTES scores for the softmax
// passes: HBM traffic ~= 290 MB (~12 us), probs touches HBM once.
//
// d_out = [ctx (4*1024*768 f32)] ++ [probs (4*12*1024*1024 f32)]
// d_ws  = Q,K,V in [B,H,S,D] fp32 (3 * 12.58 MB).
// ---------------------------------------------------------------------------

typedef __attribute__((ext_vector_type(2))) float v2f;
typedef __attribute__((ext_vector_type(8))) float v8f;

#define LSTR 66      // LDS row stride (floats): even (8B align for b64 frag
                     // loads) and 66 mod 64 == 2 -> conflict-free.
#define GSTR 130     // LDS row stride for 128-wide G/H bias panels.

// Fragment convention (ISA 7.12.2, 32-bit A 16x4): lane t holds
//   A: M = t%16, K = (t/16)*2 .. +1   (2 VGPRs)
//   B: N = t%16, K = (t/16)*2 .. +1   (B staged as Bt[N][K] in LDS)
//   C/D: M = v + (t/16)*8, N = t%16   (8 VGPRs)
#define WMMA_F32(a, b, c) \
  __builtin_amdgcn_wmma_f32_16x16x4_f32(false, (a), false, (b), (short)0, (c), false, false)

__device__ __forceinline__ v2f frag66(const float* base, int rowoff, int kk, int lane) {
  return *(const v2f*)(base + (rowoff + (lane & 15)) * LSTR + kk + ((lane >> 4) << 1));
}

// ---------------------------------------------------------------------------
// Kernel 1: fused QKV projection.  X[4096,768] @ W[768,768] + b,
// written straight into [B,H,S,D].  grid = (64 row-tiles, 12 heads, 3)
// ---------------------------------------------------------------------------
__global__ __launch_bounds__(256) void qkv_kernel(
    const float* __restrict__ X,
    const float* __restrict__ Wq, const float* __restrict__ bq,
    const float* __restrict__ Wk, const float* __restrict__ bk,
    const float* __restrict__ Wv, const float* __restrict__ bv,
    float* __restrict__ Q, float* __restrict__ K, float* __restrict__ V)
{
  const int rt = blockIdx.x;
  const int h  = blockIdx.y;
  const int wsel = blockIdx.z;
  const float* Wm = (wsel == 0) ? Wq : (wsel == 1) ? Wk : Wv;
  const float* bm = (wsel == 0) ? bq : (wsel == 1) ? bk : bv;
  float*       Om = (wsel == 0) ? Q  : (wsel == 1) ? K  : V;

  __shared__ float Xs[64 * LSTR];
  __shared__ float Wt[64 * LSTR];       // Wt[n][k]

  const int tid = threadIdx.x, lane = tid & 31, wv = tid >> 5;
  const int r0 = rt * 64, c0 = h * 64;
  const int nr  = (wv & 3) * 16;
  const int mrb = (wv >> 2) * 32;

  v8f acc0 = {}, acc1 = {};

  for (int kc = 0; kc < 12; ++kc) {
    const int k0 = kc * 64;
    __syncthreads();
    #pragma unroll
    for (int j = 0; j < 8; ++j) {
      int f2 = tid + 256 * j;
      int row = f2 >> 5, c2 = (f2 & 31) * 2;
      *(v2f*)(Xs + row * LSTR + c2) =
          *(const v2f*)(X + (size_t)(r0 + row) * 768 + k0 + c2);
    }
    #pragma unroll
    for (int j = 0; j < 8; ++j) {
      int f2 = tid + 256 * j;
      int kr = f2 >> 5, c2 = (f2 & 31) * 2;
      v2f w = *(const v2f*)(Wm + (size_t)(k0 + kr) * 768 + c0 + c2);
      Wt[(c2 + 0) * LSTR + kr] = w.x;
      Wt[(c2 + 1) * LSTR + kr] = w.y;
    }
    __syncthreads();
    #pragma unroll
    for (int kk = 0; kk < 64; kk += 4) {
      v2f bfr = frag66(Wt, nr, kk, lane);
      acc0 = WMMA_F32(frag66(Xs, mrb,      kk, lane), bfr, acc0);
      acc1 = WMMA_F32(frag66(Xs, mrb + 16, kk, lane), bfr, acc1);
    }
  }

  const int N = lane & 15, Moff = (lane >> 4) * 8;
  const int d = nr + N;
  const float bias = bm[c0 + d];
  #pragma unroll
  for (int v = 0; v < 8; ++v) {
    int g0r = r0 + mrb + v + Moff;
    int g1r = g0r + 16;
    int b0 = g0r >> 10, s0 = g0r & 1023;
    int b1 = g1r >> 10, s1 = g1r & 1023;
    Om[(((size_t)b0 * 12 + h) * 1024 + s0) * 64 + d] = acc0[v] + bias;
    Om[(((size_t)b1 * 12 + h) * 1024 + s1) * 64 + d] = acc1[v] + bias;
  }
}

// ---------------------------------------------------------------------------
// Kernel 2: fused scores + relative-position bias + softmax + P.V.
// One block per (b, h, l-tile of 64).  Three passes over the 16 r-tiles:
//   pass 1: WMMA scores -> row max (registers + shfl + LDS reduce)
//   pass 2: recompute, e = exp(s-max): row sums, write e strip,
//           ctx += e @ V via WMMA (unnormalized, flash-style)
//   pass 3: rescale strip (L2-hot) by 1/rowsum; store ctx/rowsum.
// ---------------------------------------------------------------------------
__global__ __launch_bounds__(256) void attn_kernel(
    const float* __restrict__ Q, const float* __restrict__ K,
    const float* __restrict__ V, const float* __restrict__ E,
    const float* __restrict__ mask,
    float* __restrict__ P, float* __restrict__ O)
{
  const int lt = blockIdx.x, h = blockIdx.y, b = blockIdx.z;
  const int l0 = lt * 64;

  __shared__ float smem[34176];         // 136,704 B (< 160 KB -> 2 blocks/WGP)
  float* Qs = smem;                     //  64*66 = 4224
  float* Ks = Qs + 4224;                //  64*66 = 4224
  float* Es = Ks + 4224;                // 128*66 = 8448 (pass2 alias: Ps, Vt)
  float* Gs = Es + 8448;                //  64*130 = 8320
  float* Hs = Gs + 8320;                //  64*130 = 8320
  float* rmax4  = Hs + 8320;            //  64*4
  float* rowmax = rmax4 + 256;          //  64
  float* rsum4  = rowmax + 64;          //  64*4
  float* rowinv = rsum4 + 256;          //  64 (1/rowsum)

  const int tid = threadIdx.x, lane = tid & 31, wv = tid >> 5;
  const int N = lane & 15, Moff = (lane >> 4) * 8;
  const int nrQ = (wv & 3) * 16, mrb = (wv >> 2) * 32;

  const float* Qb = Q + (((size_t)b * 12 + h) * 1024 + l0) * 64;
  const float* Kh = K + (((size_t)b * 12 + h) * 1024) * 64;
  const float* Vh = V + (((size_t)b * 12 + h) * 1024) * 64;
  float* Pb = P + (((size_t)b * 12 + h) * 1024 + l0) * 1024;

  // Q strip: persistent in LDS.
  #pragma unroll
  for (int j = 0; j < 8; ++j) {
    int f2 = tid + 256 * j;
    int row = f2 >> 5, c2 = (f2 & 31) * 2;
    *(v2f*)(Qs + row * LSTR + c2) = *(const v2f*)(Qb + row * 64 + c2);
  }

  // Compute one 64x64 score tile (QK^T + Toeplitz bias + scale + mask).
  auto tile_scores = [&](int rt, v8f& s0, v8f& s1) {
    const int r0 = rt * 64;
    const int e0 = 960 + l0 - r0;       // first E row; gather idx = 63+li-ri
    __syncthreads();                    // prior readers of Ks/Es(=Ps/Vt) done
    #pragma unroll
    for (int j = 0; j < 8; ++j) {       // K tile
      int f2 = tid + 256 * j;
      int row = f2 >> 5, c2 = (f2 & 31) * 2;
      *(v2f*)(Ks + row * LSTR + c2) = *(const v2f*)(Kh + (size_t)(r0 + row) * 64 + c2);
    }
    #pragma unroll
    for (int j = 0; j < 16; ++j) {      // 128 E rows (row 127 zeroed)
      int f2 = tid + 256 * j;
      int row = f2 >> 5, c2 = (f2 & 31) * 2;
      int er = e0 + row;
      v2f val = {0.f, 0.f};
      if (row < 127 && er >= 0 && er < 2047)
        val = *(const v2f*)(E + (size_t)er * 64 + c2);
      *(v2f*)(Es + row * LSTR + c2) = val;
    }
    __syncthreads();
    {                                   // G = Q.E^T, H = K.E^T (64x128x64)
      const int nr = wv * 16;
      v8f g0 = {}, g1 = {}, g2 = {}, g3 = {};
      v8f h0 = {}, h1 = {}, h2 = {}, h3 = {};
      #pragma unroll
      for (int kk = 0; kk < 64; kk += 4) {
        v2f bfr = frag66(Es, nr, kk, lane);
        g0 = WMMA_F32(frag66(Qs,  0, kk, lane), bfr, g0);
        g1 = WMMA_F32(frag66(Qs, 16, kk, lane), bfr, g1);
        g2 = WMMA_F32(frag66(Qs, 32, kk, lane), bfr, g2);
        g3 = WMMA_F32(frag66(Qs, 48, kk, lane), bfr, g3);
        h0 = WMMA_F32(frag66(Ks,  0, kk, lane), bfr, h0);
        h1 = WMMA_F32(frag66(Ks, 16, kk, lane), bfr, h1);
        h2 = WMMA_F32(frag66(Ks, 32, kk, lane), bfr, h2);
        h3 = WMMA_F32(frag66(Ks, 48, kk, lane), bfr, h3);
      }
      #pragma unroll
      for (int v = 0; v < 8; ++v) {
        int c = nr + N, m = v + Moff;
        Gs[(m +  0) * GSTR + c] = g0[v];  Hs[(m +  0) * GSTR + c] = h0[v];
        Gs[(m + 16) * GSTR + c] = g1[v];  Hs[(m + 16) * GSTR + c] = h1[v];
        Gs[(m + 32) * GSTR + c] = g2[v];  Hs[(m + 32) * GSTR + c] = h2[v];
        Gs[(m + 48) * GSTR + c] = g3[v];  Hs[(m + 48) * GSTR + c] = h3[v];
      }
    }
    __syncthreads();                    // all Es reads done; panels visible
    {                                   // QK^T + gather
      v8f a0 = {}, a1 = {};
      #pragma unroll
      for (int kk = 0; kk < 64; kk += 4) {
        v2f bfr = frag66(Ks, nrQ, kk, lane);
        a0 = WMMA_F32(frag66(Qs, mrb,      kk, lane), bfr, a0);
        a1 = WMMA_F32(frag66(Qs, mrb + 16, kk, lane), bfr, a1);
      }
      const int ri = nrQ + N;
      const float mval = mask[b * 1024 + r0 + ri];
      #pragma unroll
      for (int v = 0; v < 8; ++v) {
        int li0 = mrb + v + Moff, li1 = li0 + 16;
        int i0 = 63 + li0 - ri, i1 = 63 + li1 - ri;     // in [0,126]
        s0[v] = (a0[v] + Gs[li0 * GSTR + i0] + Hs[ri * GSTR + i0]) * 0.125f + mval;
        s1[v] = (a1[v] + Gs[li1 * GSTR + i1] + Hs[ri * GSTR + i1]) * 0.125f + mval;
      }
    }
  };

  // ---------------- pass 1: row max ----------------
  float pm0[8], pm1[8];
  #pragma unroll
  for (int v = 0; v < 8; ++v) { pm0[v] = -3.0e38f; pm1[v] = -3.0e38f; }
  for (int rt = 0; rt < 16; ++rt) {
    v8f s0, s1;
    tile_scores(rt, s0, s1);
    #pragma unroll
    for (int v = 0; v < 8; ++v) {
      pm0[v] = fmaxf(pm0[v], s0[v]);
      pm1[v] = fmaxf(pm1[v], s1[v]);
    }
  }
  #pragma unroll
  for (int v = 0; v < 8; ++v) {         // reduce over N (lanes xor 1,2,4,8)
    float m0 = pm0[v], m1 = pm1[v];
    #pragma unroll
    for (int mb = 1; mb <= 8; mb <<= 1) {
      m0 = fmaxf(m0, __shfl_xor(m0, mb));
      m1 = fmaxf(m1, __shfl_xor(m1, mb));
    }
    if (N == 0) {
      rmax4[(mrb + v + Moff) * 4 + (wv & 3)]      = m0;
      rmax4[(mrb + 16 + v + Moff) * 4 + (wv & 3)] = m1;
    }
  }
  __syncthreads();
  if (tid < 64)
    rowmax[tid] = fmaxf(fmaxf(rmax4[tid * 4], rmax4[tid * 4 + 1]),
                        fmaxf(rmax4[tid * 4 + 2], rmax4[tid * 4 + 3]));
  __syncthreads();

  float rm0[8], rm1[8];
  #pragma unroll
  for (int v = 0; v < 8; ++v) {
    rm0[v] = rowmax[mrb + v + Moff];
    rm1[v] = rowmax[mrb + 16 + v + Moff];
  }

  // ---------------- pass 2: sums + e strip + ctx WMMA ----------------
  float* Ps = Es;                       // alias (Es dead after gather)
  float* Vt = Es + 4224;
  v8f c0 = {}, c1 = {};                 // unnormalized ctx accumulators
  float ps0[8], ps1[8];
  #pragma unroll
  for (int v = 0; v < 8; ++v) { ps0[v] = 0.f; ps1[v] = 0.f; }

  for (int rt = 0; rt < 16; ++rt) {
    const int r0 = rt * 64;
    v8f s0, s1;
    tile_scores(rt, s0, s1);
    const int ri = nrQ + N;
    #pragma unroll
    for (int v = 0; v < 8; ++v) {
      int li0 = mrb + v + Moff, li1 = li0 + 16;
      float e0 = __expf(s0[v] - rm0[v]);
      float e1 = __expf(s1[v] - rm1[v]);
      ps0[v] += e0;  ps1[v] += e1;
      Pb[(size_t)li0 * 1024 + r0 + ri] = e0;    // unnormalized, rescaled pass 3
      Pb[(size_t)li1 * 1024 + r0 + ri] = e1;
      Ps[li0 * LSTR + ri] = e0;                 // stage e-tile as WMMA A
      Ps[li1 * LSTR + ri] = e1;
    }
    #pragma unroll
    for (int j = 0; j < 8; ++j) {               // V tile transposed: Vt[d][r]
      int f2 = tid + 256 * j;
      int kr = f2 >> 5, c2 = (f2 & 31) * 2;
      v2f w = *(const v2f*)(Vh + (size_t)(r0 + kr) * 64 + c2);
      Vt[(c2 + 0) * LSTR + kr] = w.x;
      Vt[(c2 + 1) * LSTR + kr] = w.y;
    }
    __syncthreads();
    #pragma unroll
    for (int kk = 0; kk < 64; kk += 4) {
      v2f bfr = frag66(Vt, nrQ, kk, lane);
      c0 = WMMA_F32(frag66(Ps, mrb,      kk, lane), bfr, c0);
      c1 = WMMA_F32(frag66(Ps, mrb + 16, kk, lane), bfr, c1);
    }
  }

  #pragma unroll
  for (int v = 0; v < 8; ++v) {         // row-sum reduce
    float t0 = ps0[v], t1 = ps1[v];
    #pragma unroll
    for (int mb = 1; mb <= 8; mb <<= 1) {
      t0 += __shfl_xor(t0, mb);
      t1 += __shfl_xor(t1, mb);
    }
    if (N == 0) {
      rsum4[(mrb + v + Moff) * 4 + (wv & 3)]      = t0;
      rsum4[(mrb + 16 + v + Moff) * 4 + (wv & 3)] = t1;
    }
  }
  __syncthreads();
  if (tid < 64)
    rowinv[tid] = 1.f / (rsum4[tid * 4] + rsum4[tid * 4 + 1] +
                         rsum4[tid * 4 + 2] + rsum4[tid * 4 + 3]);
  __threadfence();                      // pass-2 strip writes -> visible
  __syncthreads();

  // ---------------- pass 3: rescale strip (L2-hot) + store ctx ----------------
  #pragma unroll 4
  for (int j = 0; j < 256; ++j) {
    int f = tid + 256 * j;              // 64 rows * 1024 cols
    Pb[f] *= rowinv[f >> 10];
  }
  #pragma unroll
  for (int v = 0; v < 8; ++v) {
    int li0 = mrb + v + Moff, li1 = li0 + 16;
    O[((size_t)b * 1024 + l0 + li0) * 768 + h * 64 + nrQ + N] = c0[v] * rowinv[li0];
    O[((size_t)b * 1024 + l0 + li1) * 768 + h * 64 + nrQ + N] = c1[v] * rowinv[li1];
  }
}

// ---------------------------------------------------------------------------
extern "C" void kernel_launch(void* const* d_in, const int* in_sizes, int n_in,
                              void* d_out, int out_size, void* d_ws, size_t ws_size,
                              hipStream_t stream) {
  const float* X    = (const float*)d_in[0];   // hidden_states [4,1024,768]
  const float* mask = (const float*)d_in[1];   // attention_mask [4,1,1,1024]
  const float* Wq   = (const float*)d_in[2];
  const float* bq   = (const float*)d_in[3];
  const float* Wk   = (const float*)d_in[4];
  const float* bk   = (const float*)d_in[5];
  const float* Wv   = (const float*)d_in[6];
  const float* bv   = (const float*)d_in[7];
  const float* E    = (const float*)d_in[8];   // dist_emb [2047,64]

  float* ctx   = (float*)d_out;                              // [4,1024,768]
  float* probs = (float*)d_out + (size_t)4 * 1024 * 768;     // [4,12,1024,1024]

  const size_t qkv_elems = (size_t)4 * 12 * 1024 * 64;
  float* Qw = (float*)d_ws;
  float* Kw = Qw + qkv_elems;
  float* Vw = Kw + qkv_elems;

  qkv_kernel<<<dim3(64, 12, 3), 256, 0, stream>>>(X, Wq, bq, Wk, bk, Wv, bv,
                                                  Qw, Kw, Vw);
  attn_kernel<<<dim3(16, 12, 4), 256, 0, stream>>>(Qw, Kw, Vw, E, mask,
                                                   probs, ctx);
}